// ChebNet_37649683316998
// MI455X (gfx1250) — compile-verified
//
#include <hip/hip_runtime.h>
#include <hip/hip_bf16.h>

// ---------------- problem constants (match reference) ----------------
#define N_NODESC   100000
#define N_EDGESC   1600000
#define NUM_GRAPHSC 64
#define CH         128      // IN_CH == HID == 128
#define OUTC       10

typedef __attribute__((ext_vector_type(2))) float v2f;
typedef __attribute__((ext_vector_type(8))) float v8f;

// fp32 global atomic add via hardware global_atomic_add_f32
__device__ __forceinline__ void atomAddF(float* p, float v) {
  unsafeAtomicAdd(p, v);
}

// ---------------- utility kernels ----------------
__global__ void k_zero4(uint4* __restrict__ p, long n16) {
  long i = (long)blockIdx.x * blockDim.x + threadIdx.x;
  long stride = (long)gridDim.x * blockDim.x;
  uint4 z; z.x = 0u; z.y = 0u; z.z = 0u; z.w = 0u;
  for (; i < n16; i += stride) p[i] = z;
}

__global__ void k_deg(const int* __restrict__ src, int* deg, int n_edges) {
  int i = blockIdx.x * blockDim.x + threadIdx.x;
  if (i < n_edges) atomicAdd(&deg[src[i]], 1);
}

__global__ void k_dis(const int* __restrict__ deg, float* __restrict__ dis, int n) {
  int i = blockIdx.x * blockDim.x + threadIdx.x;
  if (i < n) {
    int d = deg[i];
    dis[i] = (d > 0) ? rsqrtf((float)d) : 0.0f;
  }
}

__global__ void k_norm(const int* __restrict__ src, const int* __restrict__ dst,
                       const float* __restrict__ dis, float* __restrict__ nrm,
                       int n_edges) {
  int i = blockIdx.x * blockDim.x + threadIdx.x;
  if (i < n_edges) nrm[i] = -dis[src[i]] * dis[dst[i]];
}

// out[dst_e] += norm_e * h[src_e]   (one wave32 per edge, lane -> 4 channels)
__global__ void k_prop(const float* __restrict__ h, const int* __restrict__ src,
                       const int* __restrict__ dst, const float* __restrict__ nrm,
                       float* out, int n_edges) {
  const int lane = threadIdx.x & 31;
  long w = (long)blockIdx.x * (blockDim.x >> 5) + (threadIdx.x >> 5);
  const long nw = (long)gridDim.x * (blockDim.x >> 5);
  for (long e = w; e < n_edges; e += nw) {
    const int s = src[e], d = dst[e];
    const float wt = nrm[e];
    const float4 hv = *(const float4*)(h + (long)s * CH + lane * 4);
    float* op = out + (long)d * CH + lane * 4;
    atomAddF(op + 0, wt * hv.x);
    atomAddF(op + 1, wt * hv.y);
    atomAddF(op + 2, wt * hv.z);
    atomAddF(op + 3, wt * hv.w);
  }
}

// t2 = 2*t2 - x0  (Chebyshev recurrence), float4 vectorized
__global__ void k_cheb2(float4* __restrict__ t2, const float4* __restrict__ x0, long n4) {
  long i = (long)blockIdx.x * blockDim.x + threadIdx.x;
  long stride = (long)gridDim.x * blockDim.x;
  for (; i < n4; i += stride) {
    float4 t = t2[i], x = x0[i];
    t.x = 2.0f * t.x - x.x; t.y = 2.0f * t.y - x.y;
    t.z = 2.0f * t.z - x.z; t.w = 2.0f * t.w - x.w;
    t2[i] = t;
  }
}

// Pack W[3][128][128] into (even,odd)-row float2 pairs: Wpk[s][p][c] = {W[s][2p][c], W[s][2p+1][c]}
__global__ void k_packW(const float* __restrict__ W, v2f* __restrict__ Wpk) {
  int idx = blockIdx.x * blockDim.x + threadIdx.x;   // 3*64*128 total
  if (idx >= 3 * 64 * CH) return;
  int c = idx & (CH - 1);
  int p = (idx >> 7) & 63;
  int s = idx >> 13;
  v2f v;
  v.x = W[((s * CH) + 2 * p) * CH + c];
  v.y = W[((s * CH) + 2 * p + 1) * CH + c];
  Wpk[idx] = v;
}

// ---------------- fused triple GEMM: out = act(b + A0@W0 + A1@W1 + A2@W2) ----------------
// One wave per 16-row x 128-col output tile. Uses V_WMMA_F32_16X16X4_F32.
// Row tiles are disjoint per wave => out may alias A0.
__global__ void k_gemm3(const float* A0, const float* A1, const float* A2,
                        const v2f* __restrict__ Wpk, const float* __restrict__ bias,
                        float* out, int row_tiles, int do_relu) {
  const int lane = threadIdx.x & 31;
  const int wave = threadIdx.x >> 5;
  const int tile = blockIdx.x * (blockDim.x >> 5) + wave;
  if (tile >= row_tiles) return;           // wave-uniform: EXEC stays all-ones for WMMA

  const int c  = lane & 15;                // output column within 16-wide sub-tile / B col
  const int hi = lane >> 4;                // lane half
  const int kb = hi << 1;                  // K sub-offset {0,2}
  const long r0 = (long)tile * 16;
  const long rowA = r0 + c;                // A row owned by this lane

  v8f acc[8];
#pragma unroll
  for (int t = 0; t < 8; ++t) {
    const float bv = bias[t * 16 + c];
#pragma unroll
    for (int v = 0; v < 8; ++v) acc[t][v] = bv;
  }

  const float* Asrc0 = A0 + rowA * CH;
  const float* Asrc1 = A1 + rowA * CH;
  const float* Asrc2 = A2 + rowA * CH;

#pragma unroll 1
  for (int s = 0; s < 3; ++s) {
    const float* A = (s == 0) ? Asrc0 : ((s == 1) ? Asrc1 : Asrc2);
    const v2f* Wb = Wpk + (long)s * 64 * CH;
#pragma unroll 4
    for (int k = 0; k < CH; k += 4) {
      // A fragment: lane holds {A[r][k+kb], A[r][k+kb+1]}  (contiguous, 8B aligned)
      const v2f a = *(const v2f*)(A + k + kb);
      const v2f* wrow = Wb + ((k + kb) >> 1) * CH;
#pragma unroll
      for (int t = 0; t < 8; ++t) {
        // B fragment: lane holds {W[k+kb][16t+c], W[k+kb+1][16t+c]} from packed pairs
        const v2f b = wrow[t * 16 + c];
        acc[t] = __builtin_amdgcn_wmma_f32_16x16x4_f32(
            /*neg_a=*/false, a, /*neg_b=*/false, b,
            /*c_mod=*/(short)0, acc[t], /*reuse_a=*/false, /*reuse_b=*/false);
      }
    }
  }

  // Epilogue: D layout -> element v of lane = out[r0 + v + 8*hi][16t + c]
#pragma unroll
  for (int t = 0; t < 8; ++t) {
#pragma unroll
    for (int v = 0; v < 8; ++v) {
      float val = acc[t][v];
      if (do_relu) val = fmaxf(val, 0.0f);
      out[(r0 + v + 8 * hi) * (long)CH + t * 16 + c] = val;
    }
  }
}

// ---------------- pooling + head ----------------
__global__ void k_count(const int* __restrict__ batch, int* cnt, int n) {
  int i = blockIdx.x * blockDim.x + threadIdx.x;
  if (i < n) atomicAdd(&cnt[batch[i]], 1);
}

// sums[batch[i]] += h[i]  (wave per node)
__global__ void k_pool(const float* __restrict__ h, const int* __restrict__ batch,
                       float* sums, int n) {
  const int lane = threadIdx.x & 31;
  long w = (long)blockIdx.x * (blockDim.x >> 5) + (threadIdx.x >> 5);
  if (w >= n) return;
  const int g = batch[w];
  const float4 hv = *(const float4*)(h + w * CH + lane * 4);
  float* op = sums + (long)g * CH + lane * 4;
  atomAddF(op + 0, hv.x);
  atomAddF(op + 1, hv.y);
  atomAddF(op + 2, hv.z);
  atomAddF(op + 3, hv.w);
}

__global__ void k_final(const float* __restrict__ sums, const int* __restrict__ cnt,
                        const float* __restrict__ lw, const float* __restrict__ lb,
                        float* __restrict__ out) {
  int idx = blockIdx.x * blockDim.x + threadIdx.x;
  if (idx >= NUM_GRAPHSC * OUTC) return;
  const int g = idx / OUTC, o = idx - g * OUTC;
  const float inv = 1.0f / fmaxf((float)cnt[g], 1.0f);
  float a = lb[o];
  for (int ch = 0; ch < CH; ++ch)
    a += (sums[g * CH + ch] * inv) * lw[ch * OUTC + o];
  out[idx] = a;
}

// ---------------- host orchestration ----------------
extern "C" void kernel_launch(void* const* d_in, const int* in_sizes, int n_in,
                              void* d_out, int out_size, void* d_ws, size_t ws_size,
                              hipStream_t stream) {
  const float* x    = (const float*)d_in[0];
  const int*   ei   = (const int*)d_in[1];
  const int*   batch= (const int*)d_in[2];
  const float* Wm[3]= {(const float*)d_in[3], (const float*)d_in[5], (const float*)d_in[7]};
  const float* bs[3]= {(const float*)d_in[4], (const float*)d_in[6], (const float*)d_in[8]};
  const float* lw   = (const float*)d_in[9];
  const float* lb   = (const float*)d_in[10];
  float* outp       = (float*)d_out;

  const int* src = ei;
  const int* dst = ei + N_EDGESC;

  // carve workspace
  char* ws = (char*)d_ws;
  size_t off = 0;
  auto carve = [&](size_t bytes) -> void* {
    void* p = ws + off; off += (bytes + 255) & ~(size_t)255; return p;
  };
  int*   deg  = (int*)  carve((size_t)N_NODESC * 4);
  float* dis  = (float*)carve((size_t)N_NODESC * 4);
  float* nrm  = (float*)carve((size_t)N_EDGESC * 4);
  float* H0   = (float*)carve((size_t)N_NODESC * CH * 4);
  float* H1   = (float*)carve((size_t)N_NODESC * CH * 4);
  float* H2   = (float*)carve((size_t)N_NODESC * CH * 4);
  v2f*   Wpk  = (v2f*)  carve((size_t)3 * 64 * CH * 8);
  float* sums = (float*)carve((size_t)NUM_GRAPHSC * CH * 4);
  int*   cnt  = (int*)  carve((size_t)NUM_GRAPHSC * 4);
  (void)ws_size; (void)n_in; (void)in_sizes; (void)out_size;

  const long featN = (long)N_NODESC * CH;        // 12.8M floats
  const int ZB = 1024;                            // blocks for zero/elemwise kernels
  const int edgeBlocks = (N_EDGESC + 7) / 8;      // one wave32 per edge, 8 waves/block
  const int rowTiles = N_NODESC / 16;             // 6250 (exact)
  const int gemmBlocks = (rowTiles + 7) / 8;

  // ---- Laplacian norm (once) ----
  k_zero4<<<256, 256, 0, stream>>>((uint4*)deg, (long)N_NODESC / 4);
  k_deg  <<<(N_EDGESC + 255) / 256, 256, 0, stream>>>(src, deg, N_EDGESC);
  k_dis  <<<(N_NODESC + 255) / 256, 256, 0, stream>>>(deg, dis, N_NODESC);
  k_norm <<<(N_EDGESC + 255) / 256, 256, 0, stream>>>(src, dst, dis, nrm, N_EDGESC);

  // ---- 3 ChebConv layers ----
  for (int l = 0; l < 3; ++l) {
    const float* Ain = (l == 0) ? x : H0;
    k_packW<<<(3 * 64 * CH + 255) / 256, 256, 0, stream>>>(Wm[l], Wpk);

    // Tx1 = P(Ain)
    k_zero4<<<ZB, 256, 0, stream>>>((uint4*)H1, featN / 4);
    k_prop <<<edgeBlocks, 256, 0, stream>>>(Ain, src, dst, nrm, H1, N_EDGESC);
    // Tx2 = 2*P(Tx1) - Ain
    k_zero4<<<ZB, 256, 0, stream>>>((uint4*)H2, featN / 4);
    k_prop <<<edgeBlocks, 256, 0, stream>>>(H1, src, dst, nrm, H2, N_EDGESC);
    k_cheb2<<<ZB, 256, 0, stream>>>((float4*)H2, (const float4*)Ain, featN / 4);

    // H0 = act(b + Ain@W0 + Tx1@W1 + Tx2@W2)   (out aliases Ain for l>0: safe, disjoint row tiles)
    k_gemm3<<<gemmBlocks, 256, 0, stream>>>(Ain, H1, H2, Wpk, bs[l], H0, rowTiles, l < 2 ? 1 : 0);
  }

  // ---- global mean pool + linear head ----
  k_zero4<<<64, 256, 0, stream>>>((uint4*)sums, (long)NUM_GRAPHSC * CH / 4);
  k_zero4<<<1, 64, 0, stream>>>((uint4*)cnt, (long)NUM_GRAPHSC / 4);
  k_count<<<(N_NODESC + 255) / 256, 256, 0, stream>>>(batch, cnt, N_NODESC);
  k_pool <<<(N_NODESC + 7) / 8, 256, 0, stream>>>(H0, batch, sums, N_NODESC);
  k_final<<<(NUM_GRAPHSC * OUTC + 127) / 128, 128, 0, stream>>>(sums, cnt, lw, lb, outp);
}